// PSA_19361712571274
// MI455X (gfx1250) — compile-verified
//
#include <hip/hip_runtime.h>
#include <math.h>

#define C_IN   256
#define C_HID  128
#define HW     (512 * 512)
#define NCHUNK 32
#define EPS    1e-5f

typedef float v2f __attribute__((ext_vector_type(2)));
typedef float v4f __attribute__((ext_vector_type(4)));
typedef float v8f __attribute__((ext_vector_type(8)));

// ---------------------------------------------------------------------------
// Kernel 1: per-channel partial sums of x.  grid=(256, NCHUNK), block=256.
// Each block reduces HW/NCHUNK = 8192 contiguous floats of one channel.
// ---------------------------------------------------------------------------
__global__ void psa_colsum(const float* __restrict__ x, float* __restrict__ partial) {
    const int c     = blockIdx.x;
    const int chunk = blockIdx.y;
    const int tid   = threadIdx.x;
    const v4f* xp = (const v4f*)(x + (size_t)c * HW + (size_t)chunk * (HW / NCHUNK));
    float s = 0.f;
#pragma unroll
    for (int i = 0; i < 8; ++i) {           // 8 * 256 threads * 4 = 8192 floats
        v4f v = xp[tid + i * 256];
        s += v.x + v.y + v.z + v.w;
    }
    __shared__ float red[256];
    red[tid] = s;
    __syncthreads();
    for (int st = 128; st > 0; st >>= 1) {
        if (tid < st) red[tid] += red[tid + st];
        __syncthreads();
    }
    if (tid == 0) partial[c * NCHUNK + chunk] = red[0];
}

// ---------------------------------------------------------------------------
// Kernel 2: all the "small" math.  1 block, 256 threads (8 wave32).
// Matvecs y = W1*sx and z = W3*y are done with V_WMMA_F32_16X16X4_F32:
//   A (16x4 f32, 2 VGPR): lane l holds rows (l&15), K = {koff, koff+1},
//     koff = 2*(l>=16)      (matches ISA 7.12.2 "32-bit A-Matrix 16x4")
//   B (4x16 f32, 2 VGPR): B[k][n] = vec[k0+k] for all n (broadcast over N),
//     so column 0 of D is the matvec partial for 16 rows.
// Each wave owns one 16-row block -> EXEC is all-ones per WMMA requirement.
// ---------------------------------------------------------------------------
__global__ void psa_gates(const float* __restrict__ partial,
                          const float* __restrict__ ch1w, const float* __restrict__ ch1b,
                          const float* __restrict__ ch3w, const float* __restrict__ ch3b,
                          const float* __restrict__ lng,  const float* __restrict__ lnb,
                          const float* __restrict__ sp1w, const float* __restrict__ sp1b,
                          const float* __restrict__ sp2w, const float* __restrict__ sp2b,
                          float* __restrict__ gout) {     // [256 g | 256 w_eff | 1 b_eff]
    __shared__ float s_sx[C_IN];
    __shared__ float s_y[C_HID];
    __shared__ float s_z[C_IN];
    __shared__ float s_red[C_IN];
    __shared__ float s_s2[C_HID];

    const int tid  = threadIdx.x;
    const int lane = tid & 31;
    const int wv   = tid >> 5;            // 0..7
    const int m    = lane & 15;
    const int koff = (lane >> 4) << 1;    // 0 or 2

    // --- sx[c] = sum_hw x[c,:] ---
    float s = 0.f;
#pragma unroll
    for (int i = 0; i < NCHUNK; ++i) s += partial[tid * NCHUNK + i];
    s_sx[tid] = s;
    __syncthreads();

#if __has_builtin(__builtin_amdgcn_wmma_f32_16x16x4_f32)
    // --- y = ch_cv1_w (128x256) @ sx + HW*b1 ---  (wave wv -> rows 16*wv..)
    {
        const int r0 = wv * 16;
        const float* wrow = ch1w + (size_t)(r0 + m) * C_IN + koff;
        v8f acc = {0.f, 0.f, 0.f, 0.f, 0.f, 0.f, 0.f, 0.f};
        for (int k0 = 0; k0 < C_IN; k0 += 4) {
            v2f a; a.x = wrow[k0];            a.y = wrow[k0 + 1];
            v2f b; b.x = s_sx[k0 + koff];     b.y = s_sx[k0 + koff + 1];
            acc = __builtin_amdgcn_wmma_f32_16x16x4_f32(false, a, false, b,
                                                        (short)0, acc, false, false);
        }
        if (m == 0) {   // lanes 0 and 16 hold column N=0: rows r0..r0+7 / r0+8..r0+15
            const int base = r0 + ((lane >> 4) << 3);
#pragma unroll
            for (int j = 0; j < 8; ++j)
                s_y[base + j] = acc[j] + (float)HW * ch1b[base + j];
        }
    }
    __syncthreads();
    // --- z = ch_cv3_w (256x128) @ y + b3 ---  (wave wv -> rows 16*wv and 128+16*wv)
    for (int half = 0; half < 2; ++half) {
        const int r0 = wv * 16 + half * 128;
        const float* wrow = ch3w + (size_t)(r0 + m) * C_HID + koff;
        v8f acc = {0.f, 0.f, 0.f, 0.f, 0.f, 0.f, 0.f, 0.f};
        for (int k0 = 0; k0 < C_HID; k0 += 4) {
            v2f a; a.x = wrow[k0];            a.y = wrow[k0 + 1];
            v2f b; b.x = s_y[k0 + koff];      b.y = s_y[k0 + koff + 1];
            acc = __builtin_amdgcn_wmma_f32_16x16x4_f32(false, a, false, b,
                                                        (short)0, acc, false, false);
        }
        if (m == 0) {
            const int base = r0 + ((lane >> 4) << 3);
#pragma unroll
            for (int j = 0; j < 8; ++j)
                s_z[base + j] = acc[j] + ch3b[base + j];
        }
        __syncthreads();
    }
#else
    // VALU fallback (keeps compile green if the builtin is absent)
    if (tid < C_HID) {
        const float* wrow = ch1w + (size_t)tid * C_IN;
        float acc = 0.f;
        for (int c = 0; c < C_IN; ++c) acc += wrow[c] * s_sx[c];
        s_y[tid] = acc + (float)HW * ch1b[tid];
    }
    __syncthreads();
    {
        const float* wrow = ch3w + (size_t)tid * C_HID;
        float acc = 0.f;
        for (int c = 0; c < C_HID; ++c) acc += wrow[c] * s_y[c];
        s_z[tid] = acc + ch3b[tid];
    }
    __syncthreads();
#endif

    // --- LayerNorm over C, sigmoid -> per-channel gate g ---
    const float zv = s_z[tid];
    s_red[tid] = zv;
    __syncthreads();
    for (int st = 128; st > 0; st >>= 1) {
        if (tid < st) s_red[tid] += s_red[tid + st];
        __syncthreads();
    }
    const float mu = s_red[0] * (1.f / C_IN);
    __syncthreads();
    const float d = zv - mu;
    s_red[tid] = d * d;
    __syncthreads();
    for (int st = 128; st > 0; st >>= 1) {
        if (tid < st) s_red[tid] += s_red[tid + st];
        __syncthreads();
    }
    const float var = s_red[0] * (1.f / C_IN);
    __syncthreads();
    const float zn  = d * rsqrtf(var + EPS) * lng[tid] + lnb[tid];
    const float gch = 1.f / (1.f + __expf(-zn));

    // --- spatial: m = sp_cv2_w @ (sx/HW) + b2 ; softmax over 128 ---
    float mval = 0.f;
    if (tid < C_HID) {
        const float* wrow = sp2w + (size_t)tid * C_IN;
        float acc = 0.f;
        for (int c = 0; c < C_IN; ++c) acc += wrow[c] * s_sx[c];
        mval = acc * (1.f / (float)HW) + sp2b[tid];
    }
    s_red[tid] = (tid < C_HID) ? mval : -3.4e38f;
    __syncthreads();
    for (int st = 128; st > 0; st >>= 1) {
        if (tid < st) s_red[tid] = fmaxf(s_red[tid], s_red[tid + st]);
        __syncthreads();
    }
    const float mx = s_red[0];
    __syncthreads();
    const float ev = (tid < C_HID) ? __expf(mval - mx) : 0.f;
    s_red[tid] = ev;
    __syncthreads();
    for (int st = 128; st > 0; st >>= 1) {
        if (tid < st) s_red[tid] += s_red[tid + st];
        __syncthreads();
    }
    const float esum = s_red[0];
    __syncthreads();
    if (tid < C_HID) s_s2[tid] = ev / esum;
    __syncthreads();

    // --- w_eff[c] = s2^T @ sp_cv1_w ; b_eff = s2 . sp_cv1_b ---
    float we = 0.f;
    for (int c_ = 0; c_ < C_HID; ++c_) we += s_s2[c_] * sp1w[(size_t)c_ * C_IN + tid];
    gout[tid]         = gch;
    gout[C_IN + tid]  = we;
    if (tid == 0) {
        float be = 0.f;
        for (int c_ = 0; c_ < C_HID; ++c_) be += s_s2[c_] * sp1b[c_];
        gout[2 * C_IN] = be;
    }
}

// ---------------------------------------------------------------------------
// Kernel 3: out[c,p] = (g[c] + sigmoid(w_eff.x[:,p] + b_eff)) * x[c,p]
// 512 blocks x 256 threads x 2 pixels.  Pass 1 (regular loads) computes the
// per-pixel dot; pass 2 re-reads the L2-resident tile with last-use NT loads
// and streams the output with NT stores so writes don't thrash L2.
// ---------------------------------------------------------------------------
__global__ void psa_main(const float* __restrict__ x, const float* __restrict__ gout,
                         float* __restrict__ out) {
    __shared__ float s_g[C_IN];
    __shared__ float s_w[C_IN];
    const int tid = threadIdx.x;
    s_g[tid] = gout[tid];
    s_w[tid] = gout[C_IN + tid];
    const float beff = gout[2 * C_IN];
    __syncthreads();

    const size_t p = (size_t)blockIdx.x * 512 + (size_t)tid * 2;
    float a0 = 0.f, a1 = 0.f;
#pragma unroll 4
    for (int c = 0; c < C_IN; ++c) {
        v2f v = *(const v2f*)(x + (size_t)c * HW + p);
        const float w = s_w[c];
        a0 = fmaf(w, v.x, a0);
        a1 = fmaf(w, v.y, a1);
    }
    const float sig0 = 1.f / (1.f + __expf(-(a0 + beff)));
    const float sig1 = 1.f / (1.f + __expf(-(a1 + beff)));

#pragma unroll 4
    for (int c = 0; c < C_IN; ++c) {
        const size_t off = (size_t)c * HW + p;
        v2f v = __builtin_nontemporal_load((const v2f*)(x + off));
        const float g = s_g[c];
        v2f o;
        o.x = (g + sig0) * v.x;
        o.y = (g + sig1) * v.y;
        __builtin_nontemporal_store(o, (v2f*)(out + off));
    }
}

// ---------------------------------------------------------------------------
extern "C" void kernel_launch(void* const* d_in, const int* in_sizes, int n_in,
                              void* d_out, int out_size, void* d_ws, size_t ws_size,
                              hipStream_t stream) {
    const float* x    = (const float*)d_in[0];
    const float* ch1w = (const float*)d_in[1];
    const float* ch1b = (const float*)d_in[2];
    // d_in[3], d_in[4] = ch_cv2_w/b: mathematically eliminated (softmax over size-1 axis)
    const float* ch3w = (const float*)d_in[5];
    const float* ch3b = (const float*)d_in[6];
    const float* lng  = (const float*)d_in[7];
    const float* lnb  = (const float*)d_in[8];
    const float* sp1w = (const float*)d_in[9];
    const float* sp1b = (const float*)d_in[10];
    const float* sp2w = (const float*)d_in[11];
    const float* sp2b = (const float*)d_in[12];
    float* out = (float*)d_out;

    float* partial = (float*)d_ws;                 // 256*NCHUNK floats
    float* gout    = partial + C_IN * NCHUNK;      // 513 floats

    psa_colsum<<<dim3(C_IN, NCHUNK), 256, 0, stream>>>(x, partial);
    psa_gates<<<1, 256, 0, stream>>>(partial, ch1w, ch1b, ch3w, ch3b,
                                     lng, lnb, sp1w, sp1b, sp2w, sp2b, gout);
    psa_main<<<HW / 512, 256, 0, stream>>>(x, gout, out);
}